// DeformableConv_28862180229379
// MI455X (gfx1250) — compile-verified
//
#include <hip/hip_runtime.h>

// DeformableConv for MI455X (gfx1250): offset conv as implicit GEMM on
// V_WMMA_F32_16X16X4_F32 (fp32-exact matrix pipe), fused with the bilinear
// gather stage. One wave32 per 16 output positions.

typedef float v2f __attribute__((ext_vector_type(2)));
typedef float v4f __attribute__((ext_vector_type(4)));
typedef float v8f __attribute__((ext_vector_type(8)));

#define FHH 5
#define FWW 5
#define DILA 2
#define NG   2
#define NOFF 100          // FH*FW*GROUPS*2
#define NTILE 7           // ceil(100/16)
#define KD   400          // FH*FW*C
#define NC   16
#define NB   2
#define IH   192
#define IW   384
#define OHH  184          // 192 - 2*4
#define OWW  376          // 384 - 2*4
#define AST  404          // LDS row stride (words): conflict-free fragment reads

// ---------------- kernel 1: S[b,y,x] = sum_c vol[b,y,x,c] ----------------
__global__ void sumc_kernel(const float* __restrict__ vol, float* __restrict__ S) {
  int i = blockIdx.x * blockDim.x + threadIdx.x;
  if (i >= NB * IH * IW) return;
  const v4f* p = (const v4f*)(vol + (size_t)i * NC);
  v4f a = p[0], b = p[1], c = p[2], d = p[3];
  v4f t = a + b + c + d;
  S[i] = t.x + t.y + t.z + t.w;
}

// ---------------- kernel 2: fused offset-GEMM (WMMA) + deformable gather ---
__global__ __launch_bounds__(32)
void deform_kernel(const float* __restrict__ vol,
                   const float* __restrict__ w_off,
                   const float* __restrict__ b_off,
                   const float* __restrict__ S,
                   float* __restrict__ out) {
  __shared__ float A[16 * AST];            // 25856 B : 16 positions x K=400
  __shared__ float Bt[(KD / 2) * 16 * 2];  // 25600 B : one 16-col B tile, (k,k+1) pairs
  __shared__ float OFF[16 * NOFF];         // 6400 B  : off[pos][n]

  const int lane = threadIdx.x;
  const int m0   = blockIdx.x * 16;        // first output position of this wave

  // ---- stage A: each (pos,tap) segment is 16 contiguous floats (64B) ----
  for (int seg = lane; seg < 16 * FHH * FWW; seg += 32) {
    int p   = seg / (FHH * FWW);
    int tap = seg % (FHH * FWW);
    int fy = tap / FWW, fx = tap % FWW;
    int m = m0 + p;
    int b = m / (OHH * OWW);
    int r = m % (OHH * OWW);
    int oy = r / OWW, ox = r % OWW;
    const float* src = vol + ((size_t)(b * IH + oy + fy * DILA) * IW + (ox + fx * DILA)) * NC;
    v4f* dst = (v4f*)(A + p * AST + tap * NC);   // p*AST*4 = p*1616 bytes, 16B aligned
    const v4f* s4 = (const v4f*)src;
    dst[0] = s4[0]; dst[1] = s4[1]; dst[2] = s4[2]; dst[3] = s4[3];
  }
  __syncthreads();

  const int col = lane & 15;   // N index of this lane (B/C fragments)
  const int khi = lane >> 4;   // 0: K sub-pair {0,1}; 1: {2,3}

  for (int nt = 0; nt < NTILE; ++nt) {
    __syncthreads();
    // ---- stage B tile: Bt[(k/2)*32 + n*2 + {0,1}] = w_off[k][nn], w_off[k+1][nn]
    for (int idx = lane; idx < (KD / 2) * 16; idx += 32) {
      int kh = idx >> 4;
      int n  = idx & 15;
      int nn = nt * 16 + n;
      float b0 = 0.f, b1 = 0.f;
      if (nn < NOFF) {
        b0 = w_off[(2 * kh)     * NOFF + nn];
        b1 = w_off[(2 * kh + 1) * NOFF + nn];
      }
      Bt[kh * 32 + n * 2 + 0] = b0;
      Bt[kh * 32 + n * 2 + 1] = b1;
    }
    __syncthreads();

    // ---- 100 x v_wmma_f32_16x16x4_f32 : off[16x16 tile] ----
    v8f acc = {};
    const float* Ab = A + (size_t)col * AST + 2 * khi;  // lane row M=col, K offset 2*khi
    #pragma unroll 4
    for (int kk = 0; kk < KD / 4; ++kk) {
      v2f av = *(const v2f*)(Ab + 4 * kk);                               // A[M][4kk+2khi ..]
      v2f bv = *(const v2f*)(Bt + (size_t)((2 * kk + khi) * 16 + col) * 2); // B pair
      acc = __builtin_amdgcn_wmma_f32_16x16x4_f32(
          false, av, false, bv, (short)0, acc, false, false);
    }

    // ---- bias + spill C fragment to OFF (C layout: VGPR v -> M = v+8*khi, N = col)
    int nn = nt * 16 + col;
    if (nn < NOFF) {
      float bia = b_off[nn];
      #pragma unroll
      for (int v = 0; v < 8; ++v) {
        int Mr = v + 8 * khi;
        OFF[Mr * NOFF + nn] = acc[v] + bia;
      }
    }
  }
  __syncthreads();

  // ---- sampling: lane -> (position p, group g); 25 taps each ----
  {
    int p = lane >> 1, g = lane & 1;
    int m = m0 + p;
    int b = m / (OHH * OWW);
    int r = m % (OHH * OWW);
    int oy = r / OWW, ox = r % OWW;
    const float* Sb = S + (size_t)b * IH * IW;
    float byv = (float)(4 + oy);                                   // by = DPAD_Y + oy
    float* ob = out + ((size_t)((b * NG + g) * OHH + oy) * OWW + ox) * (FHH * FWW);
    const float* op = OFF + p * NOFF;
    #pragma unroll
    for (int t = 0; t < FHH * FWW; ++t) {
      float dy = op[t * 4 + g];          // off[..., tap, 0, g]
      float dx = op[t * 4 + 2 + g];      // off[..., tap, 1, g]
      float kv = (float)((t / FWW) * DILA - 4);   // kys repeated by FW
      float ku = (float)((t % FWW) * DILA - 4);   // kxs tiled by FH
      float rx = kv + dy;
      float ry = ku + dx;
      int x0 = (int)rx, x1i = x0 + 1;    // trunc-toward-zero == astype(int32)
      int y0 = (int)ry, y1i = y0 + 1;
      y0  = min(max(y0,  0), IH - 1);    // clip BEFORE weights (matches reference)
      y1i = min(max(y1i, 0), IH - 1);
      x0  = min(max(x0,  0), IW - 1);
      x1i = min(max(x1i, 0), IW - 1);
      float w0 = ((float)y1i - ry)  * ((float)x1i - rx);
      float w1 = ((float)y1i - byv) * (rx - (float)x0);   // reference's quirk kept
      float w2 = (ry - (float)y0)   * ((float)x1i - rx);
      float w3 = (ry - (float)y0)   * (rx - (float)x0);
      float s00 = Sb[y0  * IW + x0 ];
      float s01 = Sb[y0  * IW + x1i];
      float s10 = Sb[y1i * IW + x0 ];
      float s11 = Sb[y1i * IW + x1i];
      ob[t] = w0 * s00 + w1 * s01 + w2 * s10 + w3 * s11;
    }
  }
}

extern "C" void kernel_launch(void* const* d_in, const int* in_sizes, int n_in,
                              void* d_out, int out_size, void* d_ws, size_t ws_size,
                              hipStream_t stream) {
  const float* vol   = (const float*)d_in[0];   // (2,192,384,16)
  const float* w_off = (const float*)d_in[1];   // (5,5,16,100)
  const float* b_off = (const float*)d_in[2];   // (100,)
  float* S   = (float*)d_ws;                    // (2,192,384) = 589,824 B scratch
  float* out = (float*)d_out;                   // (2,2,184,376,5,5)

  int npix = NB * IH * IW;
  sumc_kernel<<<(npix + 255) / 256, 256, 0, stream>>>(vol, S);

  int nwaves = (NB * OHH * OWW) / 16;           // 138368/16 = 8648, exact
  deform_kernel<<<nwaves, 32, 0, stream>>>(vol, w_off, b_off, S, out);
}